// IMUKalmanFilter_19439021982014
// MI455X (gfx1250) — compile-verified
//
#include <hip/hip_runtime.h>
#include <hip/hip_bf16.h>

typedef float v2f __attribute__((ext_vector_type(2)));
typedef float v8f __attribute__((ext_vector_type(8)));

#define DEVINL __device__ __forceinline__

// ---------------- small 3x3 helpers ----------------
DEVINL void m3mul(const float* A, const float* B, float* C) {
  for (int i = 0; i < 3; ++i)
    for (int j = 0; j < 3; ++j)
      C[i*3+j] = A[i*3+0]*B[0*3+j] + A[i*3+1]*B[1*3+j] + A[i*3+2]*B[2*3+j];
}
DEVINL void m3mv(const float* A, const float* v, float* o) {
  for (int i = 0; i < 3; ++i) o[i] = A[i*3+0]*v[0] + A[i*3+1]*v[1] + A[i*3+2]*v[2];
}
DEVINL void m3T(const float* A, float* T) {
  for (int i = 0; i < 3; ++i) for (int j = 0; j < 3; ++j) T[i*3+j] = A[j*3+i];
}
DEVINL void skew3(const float* v, float* K) {
  K[0] = 0.f;    K[1] = -v[2]; K[2] =  v[1];
  K[3] =  v[2];  K[4] = 0.f;   K[5] = -v[0];
  K[6] = -v[1];  K[7] =  v[0]; K[8] = 0.f;
}
DEVINL void exp_so3_d(const float* phi, float* R) {
  float n2 = phi[0]*phi[0] + phi[1]*phi[1] + phi[2]*phi[2];
  bool sm = n2 < 1e-16f;
  float n  = sqrtf(sm ? 1.f : n2);
  float A  = sm ? 1.f : (sinf(n) / n);
  float B  = sm ? 0.5f : ((1.f - cosf(n)) / n2);
  float K[9], K2[9];
  skew3(phi, K);
  m3mul(K, K, K2);
  for (int i = 0; i < 9; ++i) R[i] = A * K[i] + B * K2[i];
  R[0] += 1.f; R[4] += 1.f; R[8] += 1.f;
}
DEVINL void log_so3_d(const float* C, float* o) {
  float tr = C[0] + C[4] + C[8];
  float cs = (tr - 1.f) * 0.5f;
  cs = fminf(fmaxf(cs, -1.f + 1e-7f), 1.f - 1e-7f);
  float pn = acosf(cs);
  float s  = sinf(pn);
  float u0 = C[7] - C[5], u1 = C[2] - C[6], u2 = C[3] - C[1];
  float f  = (s > 1e-6f) ? pn / (2.f * s) : 0.5f;
  o[0] = f * u0; o[1] = f * u1; o[2] = f * u2;
}
DEVINL void jlinv_d(const float* phi, float* J) {
  float n2 = phi[0]*phi[0] + phi[1]*phi[1] + phi[2]*phi[2];
  bool sm = n2 < 1e-12f;
  float n = sqrtf(sm ? 1.f : n2);
  float a[3] = { phi[0]/(sm?1.f:n), phi[1]/(sm?1.f:n), phi[2]/(sm?1.f:n) };
  float half = 0.5f * n;
  float cot  = cosf(half) / (sm ? 1.f : sinf(half));
  float hc   = half * cot;
  float K[9]; skew3(a, K);
  float Kp[9]; skew3(phi, Kp);
  for (int i = 0; i < 3; ++i)
    for (int j = 0; j < 3; ++j) {
      float big = hc * (i==j ? 1.f : 0.f) + (1.f - hc) * a[i]*a[j] - half * K[i*3+j];
      float sml = (i==j ? 1.f : 0.f) - 0.5f * Kp[i*3+j];
      J[i*3+j] = sm ? sml : big;
    }
}

// ---------------- 32x32 f32 WMMA matmul through LDS ----------------
DEVINL v8f wmma4(v2f a, v2f b, v8f c) {
  // V_WMMA_F32_16X16X4_F32 : D = A(16x4) x B(4x16) + C(16x16), all f32
  return __builtin_amdgcn_wmma_f32_16x16x4_f32(
      false, a, false, b, (short)0, c, false, false);
}

// D = A @ B (or A @ B^T), optionally + Cin, times scale. All row-major 32x32 LDS.
__device__ void mm32(const float* __restrict__ A, const float* __restrict__ B,
                     const float* __restrict__ Cin, float* __restrict__ D,
                     bool transB, float scale, int lane) {
  const int n = lane & 15;     // column within 16x16 tile / A row
  const int h = lane >> 4;     // half select (K pairs / row+8)
  #pragma unroll
  for (int bi = 0; bi < 2; ++bi) {
    #pragma unroll
    for (int bj = 0; bj < 2; ++bj) {
      v8f acc = {};
      if (Cin) {
        #pragma unroll
        for (int vv = 0; vv < 8; ++vv)
          acc[vv] = Cin[(16*bi + vv + 8*h)*32 + 16*bj + n];
      }
      #pragma unroll
      for (int kb = 0; kb < 8; ++kb) {
        const int k0 = 4*kb + 2*h;
        v2f a, b;
        a[0] = A[(16*bi + n)*32 + k0];
        a[1] = A[(16*bi + n)*32 + k0 + 1];
        if (!transB) {
          b[0] = B[(k0    )*32 + 16*bj + n];
          b[1] = B[(k0 + 1)*32 + 16*bj + n];
        } else {
          b[0] = B[(16*bj + n)*32 + k0];
          b[1] = B[(16*bj + n)*32 + k0 + 1];
        }
        acc = wmma4(a, b, acc);
      }
      #pragma unroll
      for (int vv = 0; vv < 8; ++vv)
        D[(16*bi + vv + 8*h)*32 + 16*bj + n] = acc[vv] * scale;
    }
  }
}

// =================== Kernel 1: rotation / v_acc prefix scan ===================
#define K1T 512
__global__ __launch_bounds__(K1T)
void prefix_kernel(const float* __restrict__ imu, const float* __restrict__ prev_state,
                   float* __restrict__ Cpre, float* __restrict__ vaccb,
                   float* __restrict__ finals, int N) {
  __shared__ float sR[K1T * 9];
  __shared__ float sV[K1T * 3];
  __shared__ float sS[K1T * 3];
  const int t = threadIdx.x;
  const int S = (N + K1T - 1) / K1T;
  const int i0 = t * S;
  const int i1 = (i0 + S < N) ? (i0 + S) : N;
  float bw[3], ba[3];
  for (int k = 0; k < 3; ++k) { bw[k] = prev_state[18+k]; ba[k] = prev_state[21+k]; }

  // pass 1: thread-local rotation product
  float R[9] = {1,0,0, 0,1,0, 0,0,1};
  for (int i = i0; i < i1; ++i) {
    float dt = imu[(i+1)*7] - imu[i*7];
    float dphi[3];
    for (int k = 0; k < 3; ++k) dphi[k] = dt * (imu[i*7+1+k] - bw[k]);
    float ew[9]; exp_so3_d(dphi, ew);
    float Rn[9]; m3mul(R, ew, Rn);
    for (int k = 0; k < 9; ++k) R[k] = Rn[k];
  }
  for (int k = 0; k < 9; ++k) sR[t*9+k] = R[k];
  __syncthreads();
  if (t == 0) {  // serial exclusive scan over 512 thread totals (in place)
    float carry[9] = {1,0,0, 0,1,0, 0,0,1};
    for (int tt = 0; tt < K1T; ++tt) {
      float nxt[9]; m3mul(carry, &sR[tt*9], nxt);
      for (int k = 0; k < 9; ++k) { sR[tt*9+k] = carry[k]; carry[k] = nxt[k]; }
    }
    for (int k = 0; k < 9; ++k) finals[k] = carry[k];   // C_final
    finals[15] = imu[N*7] - imu[0];                     // t_acc total
  }
  __syncthreads();

  // pass 2: store per-step exclusive prefix rotations; thread-local dv/dr sums
  float C[9]; for (int k = 0; k < 9; ++k) C[k] = sR[t*9+k];
  float sv[3] = {0,0,0}, sr[3] = {0,0,0};
  for (int i = i0; i < i1; ++i) {
    for (int k = 0; k < 9; ++k) Cpre[(size_t)i*9+k] = C[k];
    float dt = imu[(i+1)*7] - imu[i*7];
    float a[3]; for (int k = 0; k < 3; ++k) a[k] = imu[i*7+4+k] - ba[k];
    float ca[3]; m3mv(C, a, ca);
    for (int k = 0; k < 3; ++k) { sv[k] += dt*ca[k]; sr[k] += 0.5f*dt*dt*ca[k]; }
    float dphi[3]; for (int k = 0; k < 3; ++k) dphi[k] = dt * (imu[i*7+1+k] - bw[k]);
    float ew[9]; exp_so3_d(dphi, ew);
    float Cn[9]; m3mul(C, ew, Cn);
    for (int k = 0; k < 9; ++k) C[k] = Cn[k];
  }
  for (int k = 0; k < 3; ++k) { sV[t*3+k] = sv[k]; sS[t*3+k] = sr[k]; }
  __syncthreads();
  if (t == 0) {
    float vc[3] = {0,0,0}, rt[3] = {0,0,0};
    for (int tt = 0; tt < K1T; ++tt)
      for (int k = 0; k < 3; ++k) {
        float nv = vc[k] + sV[tt*3+k];
        sV[tt*3+k] = vc[k]; vc[k] = nv;
        rt[k] += sS[tt*3+k];
      }
    for (int k = 0; k < 3; ++k) { finals[9+k] = vc[k]; finals[12+k] = rt[k]; }
  }
  __syncthreads();

  // pass 3: per-step exclusive prefix v_acc
  float va[3]; for (int k = 0; k < 3; ++k) va[k] = sV[t*3+k];
  for (int i = i0; i < i1; ++i) {
    for (int k = 0; k < 3; ++k) vaccb[(size_t)i*3+k] = va[k];
    float dt = imu[(i+1)*7] - imu[i*7];
    float a[3]; for (int k = 0; k < 3; ++k) a[k] = imu[i*7+4+k] - ba[k];
    float Cl[9]; for (int k = 0; k < 9; ++k) Cl[k] = Cpre[(size_t)i*9+k];
    float ca[3]; m3mv(Cl, a, ca);
    for (int k = 0; k < 3; ++k) va[k] += dt*ca[k];
  }
}

// =================== Kernel 2: per-chunk (Phi,Q) composition via WMMA ===================
__global__ __launch_bounds__(32)
void chunk_compose_kernel(const float* __restrict__ imu, const float* __restrict__ prev_state,
                          const float* __restrict__ imu_noise,
                          const float* __restrict__ Cpre, const float* __restrict__ vaccb,
                          float* __restrict__ chunkPhi, float* __restrict__ chunkQ,
                          int N, int M) {
  __shared__ float sPhi[1024], sG[1024], sNw[1024], sW[1024],
                   sT1[1024], sQi[1024], sPA[1024], sQA[1024];
  const int lane = threadIdx.x;
  const int c = blockIdx.x;
  for (int idx = lane; idx < 1024; idx += 32) {
    int rr = idx >> 5, cc = idx & 31;
    sNw[idx] = (rr < 12 && cc < 12) ? imu_noise[rr*12+cc] : 0.f;
    sPA[idx] = (rr == cc) ? 1.f : 0.f;  // padded identity accumulator
    sQA[idx] = 0.f;
  }
  float g[3], vk[3], bw[3], ba[3];
  for (int k = 0; k < 3; ++k) {
    g[k]  = prev_state[k];     vk[k] = prev_state[15+k];
    bw[k] = prev_state[18+k];  ba[k] = prev_state[21+k];
  }
  const float t0 = imu[0];
  __syncthreads();

  const int i0 = c*M;
  int i1 = (c+1)*M; if (i1 > N) i1 = N;
  for (int i = i0; i < i1; ++i) {
    // ---- per-step scalars (all lanes redundantly) ----
    float ti  = imu[i*7];
    float dt  = imu[(i+1)*7] - ti;
    float tac = ti - t0;
    float w[3];  for (int k = 0; k < 3; ++k) w[k] = imu[i*7+1+k] - bw[k];
    float C[9];  for (int k = 0; k < 9; ++k) C[k] = Cpre[(size_t)i*9+k];
    float Ct[9]; m3T(C, Ct);
    float va[3]; for (int k = 0; k < 3; ++k) va[k] = vaccb[(size_t)i*3+k];
    float tv[3]; for (int k = 0; k < 3; ++k) tv[k] = vk[k] - g[k]*tac + va[k];
    float vv[3]; m3mv(Ct, tv, vv);
    float wsk[9];  skew3(w, wsk);
    float vsk[9];  skew3(vv, vsk);
    float Cvsk[9]; m3mul(C, vsk, Cvsk);
    float ctg[3];  m3mv(Ct, g, ctg);
    float ctgsk[9]; skew3(ctg, ctgsk);
    float dphi[3] = { dt*w[0], dt*w[1], dt*w[2] };
    float ew[9]; exp_so3_d(dphi, ew);
    const float fdt = dt + 0.5f*dt*dt;

    // ---- build Phi_i row (lane = row) and G row into LDS ----
    const int r = lane;
    const int br = r / 3, ii = r % 3;
    for (int cc = 0; cc < 32; ++cc) {
      float p;
      if (r < 18 && cc < 18) {
        int bc = cc / 3, jj = cc % 3;
        float f = 0.f;
        if (br == 1)      { if (bc == 1) f = -wsk[ii*3+jj];
                            else if (bc == 4) f = -(ii==jj ? 1.f : 0.f); }
        else if (br == 2) { if (bc == 1) f = -Cvsk[ii*3+jj];
                            else if (bc == 3) f = C[ii*3+jj]; }
        else if (br == 3) { if (bc == 0) f = -Ct[ii*3+jj];
                            else if (bc == 1) f = -ctgsk[ii*3+jj];
                            else if (bc == 3) f = -wsk[ii*3+jj];
                            else if (bc == 4) f = -vsk[ii*3+jj];
                            else if (bc == 5) f = -(ii==jj ? 1.f : 0.f); }
        p = (r == cc ? 1.f : 0.f) + f * fdt;
        if (br == 1 && bc == 1) p = ew[ii*3+jj];
        if (br == 3 && bc == 3) p = ew[ii*3+jj];
      } else {
        p = (r == cc) ? 1.f : 0.f;   // identity padding keeps composition exact
      }
      sPhi[r*32 + cc] = p;
      float gg = 0.f;
      if (r < 18 && cc < 12) {
        int bc = cc / 3, jj = cc % 3;
        if (br == 1 && bc == 0) gg = -(ii==jj ? 1.f : 0.f);
        else if (br == 3 && bc == 0) gg = -vsk[ii*3+jj];
        else if (br == 3 && bc == 2) gg = -(ii==jj ? 1.f : 0.f);
        else if ((br == 4 || br == 5) && bc == 1) gg = (ii==jj ? 1.f : 0.f);
      }
      sG[r*32 + cc] = gg;
    }
    __syncthreads();

    // ---- Q_i = (Phi G Nw G^T Phi^T) * dt ----
    mm32(sG,   sNw,  nullptr, sT1, false, 1.f, lane); __syncthreads();
    mm32(sT1,  sG,   nullptr, sW,  true,  1.f, lane); __syncthreads();
    mm32(sPhi, sW,   nullptr, sT1, false, 1.f, lane); __syncthreads();
    mm32(sT1,  sPhi, nullptr, sQi, true,  dt,  lane); __syncthreads();
    // ---- compose: Qacc = Phi Qacc Phi^T + Qi ; PhiAcc = Phi PhiAcc ----
    mm32(sPhi, sQA,  nullptr, sT1, false, 1.f, lane); __syncthreads();
    mm32(sT1,  sPhi, sQi,    sQA, true,  1.f, lane);  __syncthreads();
    mm32(sPhi, sPA,  nullptr, sT1, false, 1.f, lane); __syncthreads();
    for (int idx = lane; idx < 1024; idx += 32) sPA[idx] = sT1[idx];
    __syncthreads();
  }
  for (int idx = lane; idx < 1024; idx += 32) {
    chunkPhi[(size_t)c*1024 + idx] = sPA[idx];
    chunkQ  [(size_t)c*1024 + idx] = sQA[idx];
  }
}

// =================== Kernel 3: compose chunks onto (I, P0); update+composition ===================
__global__ __launch_bounds__(32)
void final_kernel(const float* __restrict__ prev_pose, const float* __restrict__ prev_state,
                  const float* __restrict__ prev_covar, const float* __restrict__ vis_meas,
                  const float* __restrict__ vis_covar, const float* __restrict__ C_cal,
                  const float* __restrict__ r_cal, const float* __restrict__ finals,
                  const float* __restrict__ chunkPhi, const float* __restrict__ chunkQ,
                  float* __restrict__ out, int NC) {
  __shared__ float sPhi[1024], sQi[1024], sQA[1024], sT1[1024];
  const int lane = threadIdx.x;
  for (int idx = lane; idx < 1024; idx += 32) {
    int rr = idx >> 5, cc = idx & 31;
    sQA[idx] = (rr < 18 && cc < 18) ? prev_covar[rr*18+cc] : 0.f;  // Acc = (I, P0)
  }
  __syncthreads();
  for (int c = 0; c < NC; ++c) {
    for (int idx = lane; idx < 1024; idx += 32) {
      sPhi[idx] = chunkPhi[(size_t)c*1024 + idx];
      sQi[idx]  = chunkQ  [(size_t)c*1024 + idx];
    }
    __syncthreads();
    mm32(sPhi, sQA, nullptr, sT1, false, 1.f, lane); __syncthreads();
    mm32(sT1, sPhi, sQi,    sQA, true,  1.f, lane);  __syncthreads();
  }
  if (lane == 0) {
    // predicted covariance
    float P[324];
    for (int r = 0; r < 18; ++r)
      for (int c = 0; c < 18; ++c) P[r*18+c] = sQA[r*32+c];
    // predicted state
    float g[3], vk[3], bw[3], ba[3];
    for (int k = 0; k < 3; ++k) {
      g[k] = prev_state[k]; vk[k] = prev_state[15+k];
      bw[k] = prev_state[18+k]; ba[k] = prev_state[21+k];
    }
    float Cf[9]; for (int k = 0; k < 9; ++k) Cf[k] = finals[k];
    float vtot[3], rtot[3];
    for (int k = 0; k < 3; ++k) { vtot[k] = finals[9+k]; rtot[k] = finals[12+k]; }
    float tac = finals[15];
    float r_new[3];
    for (int k = 0; k < 3; ++k) r_new[k] = vk[k]*tac - 0.5f*g[k]*tac*tac + rtot[k];
    float Cft[9]; m3T(Cf, Cft);
    float tv[3]; for (int k = 0; k < 3; ++k) tv[k] = vk[k] - g[k]*tac + vtot[k];
    float v_new[3]; m3mv(Cft, tv, v_new);

    // ---------- update ----------
    float Cc[9], Cct[9], rc[3];
    for (int k = 0; k < 9; ++k) Cc[k] = C_cal[k];
    m3T(Cc, Cct);
    for (int k = 0; k < 3; ++k) rc[k] = r_cal[k];
    float evm[9]; exp_so3_d(&vis_meas[3], evm);
    float CcCp[9]; m3mul(Cct, Cf, CcCp);
    float t2[9], t3[9];
    m3mul(evm, CcCp, t2); m3mul(t2, Cc, t3);
    float res_rot[3]; log_so3_d(t3, res_rot);
    float cr[3]; m3mv(CcCp, rc, cr);
    float rmrc[3]; for (int k = 0; k < 3; ++k) rmrc[k] = r_new[k] - rc[k];
    float ctr[3]; m3mv(Cct, rmrc, ctr);
    float res_t[3]; for (int k = 0; k < 3; ++k) res_t[k] = vis_meas[k] - cr[k] - ctr[k];
    float H[6*18]; for (int k = 0; k < 108; ++k) H[k] = 0.f;
    float nres[3] = { -res_rot[0], -res_rot[1], -res_rot[2] };
    float JL[9]; jlinv_d(nres, JL);
    float Hrot[9]; m3mul(JL, CcCp, Hrot);
    float rsk[9]; skew3(r_new, rsk);
    float Htr[9]; m3mul(CcCp, rsk, Htr);
    for (int i = 0; i < 3; ++i)
      for (int j = 0; j < 3; ++j) {
        H[i*18 + 3 + j]     = Hrot[i*3+j];
        H[(3+i)*18 + 3 + j] = Htr[i*3+j];
        H[(3+i)*18 + 6 + j] = -Cct[i*3+j];
      }
    float HP[6*18];
    for (int i = 0; i < 6; ++i)
      for (int c = 0; c < 18; ++c) {
        float s = 0.f;
        for (int k = 0; k < 18; ++k) s += H[i*18+k]*P[k*18+c];
        HP[i*18+c] = s;
      }
    float Sm[36];
    for (int i = 0; i < 6; ++i)
      for (int j = 0; j < 6; ++j) {
        float s = 0.f;
        for (int c = 0; c < 18; ++c) s += HP[i*18+c]*H[j*18+c];
        Sm[i*6+j] = s + vis_covar[i*6+j];
      }
    float Rhs[6*18];   // H @ P^T
    for (int i = 0; i < 6; ++i)
      for (int c = 0; c < 18; ++c) {
        float s = 0.f;
        for (int k = 0; k < 18; ++k) s += H[i*18+k]*P[c*18+k];
        Rhs[i*18+c] = s;
      }
    // Gauss-Jordan solve Sm X = Rhs (X = 6x18), K = X^T
    for (int col = 0; col < 6; ++col) {
      int p = col; float best = fabsf(Sm[col*6+col]);
      for (int r2 = col+1; r2 < 6; ++r2) {
        float vbest = fabsf(Sm[r2*6+col]);
        if (vbest > best) { best = vbest; p = r2; }
      }
      if (p != col) {
        for (int j = 0; j < 6; ++j)  { float tt = Sm[col*6+j];  Sm[col*6+j]  = Sm[p*6+j];  Sm[p*6+j]  = tt; }
        for (int j = 0; j < 18; ++j) { float tt = Rhs[col*18+j]; Rhs[col*18+j] = Rhs[p*18+j]; Rhs[p*18+j] = tt; }
      }
      float inv = 1.f / Sm[col*6+col];
      for (int j = 0; j < 6; ++j)  Sm[col*6+j]  *= inv;
      for (int j = 0; j < 18; ++j) Rhs[col*18+j] *= inv;
      for (int r2 = 0; r2 < 6; ++r2) if (r2 != col) {
        float f = Sm[r2*6+col];
        for (int j = 0; j < 6; ++j)  Sm[r2*6+j]  -= f*Sm[col*6+j];
        for (int j = 0; j < 18; ++j) Rhs[r2*18+j] -= f*Rhs[col*18+j];
      }
    }
    float resid[6] = { res_rot[0], res_rot[1], res_rot[2], res_t[0], res_t[1], res_t[2] };
    float e[18];
    for (int r = 0; r < 18; ++r) {
      float s = 0.f;
      for (int i = 0; i < 6; ++i) s += Rhs[i*18+r]*resid[i];   // K[r][i] = X[i][r]
      e[r] = s;
    }
    float ImKH[324];
    for (int r = 0; r < 18; ++r)
      for (int c = 0; c < 18; ++c) {
        float s = 0.f;
        for (int i = 0; i < 6; ++i) s += Rhs[i*18+r]*H[i*18+c];
        ImKH[r*18+c] = (r == c ? 1.f : 0.f) - s;
      }
    float Tm[324];
    for (int r = 0; r < 18; ++r)
      for (int c = 0; c < 18; ++c) {
        float s = 0.f;
        for (int k = 0; k < 18; ++k) s += ImKH[r*18+k]*P[k*18+c];
        Tm[r*18+c] = s;
      }
    float EC[324];
    for (int r = 0; r < 18; ++r)
      for (int c = 0; c < 18; ++c) {
        float s = 0.f;
        for (int k = 0; k < 18; ++k) s += Tm[r*18+k]*ImKH[c*18+k];
        EC[r*18+c] = s;
      }
    float KR[18*6];
    for (int r = 0; r < 18; ++r)
      for (int i = 0; i < 6; ++i) {
        float s = 0.f;
        for (int j = 0; j < 6; ++j) s += Rhs[j*18+r]*vis_covar[j*6+i];
        KR[r*6+i] = s;
      }
    for (int r = 0; r < 18; ++r)
      for (int c = 0; c < 18; ++c) {
        float s = 0.f;
        for (int i = 0; i < 6; ++i) s += KR[r*6+i]*Rhs[i*18+c];
        EC[r*18+c] += s;
      }
    // est_state
    float ge[3], re[3], ve[3], bwe[3], bae[3];
    for (int k = 0; k < 3; ++k) {
      ge[k]  = g[k]    + e[k];
      re[k]  = r_new[k] + e[6+k];
      ve[k]  = v_new[k] + e[9+k];
      bwe[k] = bw[k]   + e[12+k];
      bae[k] = ba[k]   + e[15+k];
    }
    float ee[9]; exp_so3_d(&e[3], ee);
    float Ce[9]; m3mul(Cf, ee, Ce);

    // ---------- composition ----------
    float Cet[9]; m3T(Ce, Cet);
    float pose[16];
    for (int i = 0; i < 3; ++i)
      for (int j = 0; j < 3; ++j) {
        float s = 0.f;
        for (int k = 0; k < 3; ++k) s += Cet[i*3+k]*prev_pose[k*4+j];
        pose[i*4+j] = s;
      }
    float pr[3]; for (int k = 0; k < 3; ++k) pr[k] = prev_pose[k*4+3] - re[k];
    float pt[3]; m3mv(Cet, pr, pt);
    for (int i = 0; i < 3; ++i) pose[i*4+3] = pt[i];
    pose[12] = 0.f; pose[13] = 0.f; pose[14] = 0.f; pose[15] = 1.f;
    float ng[3]; m3mv(Cet, ge, ng);
    float nst[24];
    for (int k = 0; k < 3; ++k) nst[k] = ng[k];
    for (int k = 0; k < 9; ++k) nst[3+k] = (k % 4 == 0) ? 1.f : 0.f;  // I3
    for (int k = 0; k < 3; ++k) { nst[12+k] = 0.f; nst[15+k] = ve[k]; nst[18+k] = bwe[k]; nst[21+k] = bae[k]; }
    float U[324]; for (int k = 0; k < 324; ++k) U[k] = 0.f;
    float ngsk[9]; skew3(ng, ngsk);
    for (int i = 0; i < 3; ++i)
      for (int j = 0; j < 3; ++j) {
        U[i*18 + j]     = Cet[i*3+j];
        U[i*18 + 3 + j] = ngsk[i*3+j];
      }
    float T2[324];
    for (int r = 0; r < 18; ++r)
      for (int c = 0; c < 18; ++c) {
        float s = 0.f;
        for (int k = 0; k < 18; ++k) s += U[r*18+k]*EC[k*18+c];
        T2[r*18+c] = s;
      }
    // outputs: pose(16) | new_state(24) | new_covar(324)
    for (int k = 0; k < 16; ++k) out[k] = pose[k];
    for (int k = 0; k < 24; ++k) out[16+k] = nst[k];
    for (int r = 0; r < 18; ++r)
      for (int c = 0; c < 18; ++c) {
        float s = 0.f;
        for (int k = 0; k < 18; ++k) s += T2[r*18+k]*U[c*18+k];
        out[40 + r*18 + c] = s;
      }
  }
}

// =================== host launcher ===================
extern "C" void kernel_launch(void* const* d_in, const int* in_sizes, int n_in,
                              void* d_out, int out_size, void* d_ws, size_t ws_size,
                              hipStream_t stream) {
  const float* imu        = (const float*)d_in[0];
  const float* prev_pose  = (const float*)d_in[1];
  const float* prev_state = (const float*)d_in[2];
  const float* prev_covar = (const float*)d_in[3];
  const float* vis_meas   = (const float*)d_in[4];
  const float* vis_covar  = (const float*)d_in[5];
  const float* imu_noise  = (const float*)d_in[6];
  const float* C_cal      = (const float*)d_in[7];
  const float* r_cal      = (const float*)d_in[8];
  (void)n_in; (void)out_size; (void)ws_size;

  const int T = in_sizes[0] / 7;
  const int N = T - 1;
  const int NC = 256;
  const int M = (N + NC - 1) / NC;

  float* ws       = (float*)d_ws;
  float* Cpre     = ws;                               // N*9
  float* vaccb    = Cpre + (size_t)N * 9;             // N*3
  float* finals   = vaccb + (size_t)N * 3;            // 16
  float* chunkPhi = finals + 16;                      // NC*1024
  float* chunkQ   = chunkPhi + (size_t)NC * 1024;     // NC*1024

  prefix_kernel<<<1, K1T, 0, stream>>>(imu, prev_state, Cpre, vaccb, finals, N);
  chunk_compose_kernel<<<NC, 32, 0, stream>>>(imu, prev_state, imu_noise,
                                              Cpre, vaccb, chunkPhi, chunkQ, N, M);
  final_kernel<<<1, 32, 0, stream>>>(prev_pose, prev_state, prev_covar, vis_meas,
                                     vis_covar, C_cal, r_cal, finals,
                                     chunkPhi, chunkQ, (float*)d_out, NC);
}